// GuandanModel_52596169507138
// MI455X (gfx1250) — compile-verified
//
#include <hip/hip_runtime.h>
#include <math.h>

// ---------------------------------------------------------------------------
// Guandan model forward on gfx1250 (MI455X). All matmuls go through
// V_WMMA_F32_16X16X4_F32 (exact fp32 on the matrix pipes). Workload is
// bandwidth-bound (~0.42 TFLOP vs ~4 GB traffic), so fp32 WMMA is the
// right precision: exact numerics, matrix-pipe issue, weights resident in L2.
//
// GEMM: one 128-thread block (4 waves) computes a 16x64 strip of C.
// All 4 waves fetch the identical A 16x4 slab each K-step (coalesces in
// WGP$/L2); pointer-increment addressing keeps the inner loop to
// 1x global_load_b64 (A) + 2x global_load_b32 (B) + 1x v_wmma per step.
// ---------------------------------------------------------------------------

typedef float v2f __attribute__((ext_vector_type(2)));
typedef float v8f __attribute__((ext_vector_type(8)));

#define LN_EPS 1e-5f
#define POST_NONE  0
#define POST_LEAKY 1
#define POST_GELU  2

// GEMM modes
#define GM_BIAS     0   // C = acc + bias
#define GM_LEAKY    1   // C = leaky(acc + bias)
#define GM_RESID    2   // C = acc + bias + resid
#define GM_SCALEACC 3   // C += rowscale[r] * (acc + bias)

// ---------------------------------------------------------------------------
// Generic WMMA fp32 GEMM: C[M,N] = A[M,K] @ W[K,N] (+bias, epilogue).
// A layout (16x4 f32, ISA 7.12.2): lane = M%16 (+16 for K-half),
//   VGPR v holds K = 2*(lane>=16)+v.  B: N = lane%16, K = 2*(lane>=16)+v.
// C/D: N = lane%16, M = v + 8*(lane>=16).
// Grid: (N/64, M/16), block 128 (4 waves, each one 16x16 tile).
// ---------------------------------------------------------------------------
__global__ void wmma_gemm_kernel(const float* __restrict__ A, int lda,
                                 const float* __restrict__ W, int ldb,
                                 const float* __restrict__ bias,
                                 const float* __restrict__ resid, int ldr,
                                 const float* __restrict__ rowscale, int rs_stride,
                                 float* __restrict__ C, int ldc,
                                 int K, int mode) {
  const int lane = threadIdx.x & 31;
  const int wave = threadIdx.x >> 5;
  const int half = lane >> 4;
  const int l16  = lane & 15;
  const int m0 = blockIdx.y << 4;
  const int n0 = (blockIdx.x << 6) + (wave << 4);

  const float* ap = A + (size_t)(m0 + l16) * lda + (half << 1);
  const float* bp = W + (size_t)(half << 1) * ldb + n0 + l16;
  const size_t ldbs = (size_t)ldb;
  const size_t ldb4 = ldbs * 4;

  v8f acc = {};
  for (int k = 0; k < K; k += 4) {
    v2f av = *(const v2f*)ap;            // global_load_b64 (8B-aligned)
    v2f bv;
    bv.x = bp[0];
    bv.y = bp[ldbs];
    acc = __builtin_amdgcn_wmma_f32_16x16x4_f32(false, av, false, bv,
                                                (short)0, acc, false, false);
    ap += 4;
    bp += ldb4;
  }

  const int col = n0 + l16;
  const float bval = bias ? bias[col] : 0.0f;
#pragma unroll
  for (int v = 0; v < 8; ++v) {
    int r = m0 + v + (half << 3);
    float val = acc[v] + bval;
    if (mode == GM_LEAKY)      val = (val >= 0.0f) ? val : 0.01f * val;
    else if (mode == GM_RESID) val += resid[(size_t)r * ldr + col];
    size_t ci = (size_t)r * ldc + col;
    if (mode == GM_SCALEACC) C[ci] += rowscale[(size_t)r * rs_stride] * val;
    else                     C[ci] = val;
  }
}

// ---------------------------------------------------------------------------
// Row LayerNorm (+ optional residual input, + optional post-op).
// One 256-thread block per row.
// ---------------------------------------------------------------------------
__global__ void ln_kernel(const float* __restrict__ X,
                          const float* __restrict__ R_,
                          const float* __restrict__ g,
                          const float* __restrict__ b,
                          float* __restrict__ Y, int N, int post) {
  const int row = blockIdx.x;
  const int tid = threadIdx.x;
  __shared__ float s1[256], s2[256];
  __shared__ float smean, srstd;

  float sum = 0.0f, sq = 0.0f;
  for (int i = tid; i < N; i += 256) {
    float v = X[(size_t)row * N + i];
    if (R_) v += R_[(size_t)row * N + i];
    sum += v; sq += v * v;
  }
  s1[tid] = sum; s2[tid] = sq;
  __syncthreads();
  for (int s = 128; s > 0; s >>= 1) {
    if (tid < s) { s1[tid] += s1[tid + s]; s2[tid] += s2[tid + s]; }
    __syncthreads();
  }
  if (tid == 0) {
    float mean = s1[0] / N;
    float var = fmaxf(s2[0] / N - mean * mean, 0.0f);
    smean = mean;
    srstd = rsqrtf(var + LN_EPS);
  }
  __syncthreads();
  const float mean = smean, rstd = srstd;
  for (int i = tid; i < N; i += 256) {
    float v = X[(size_t)row * N + i];
    if (R_) v += R_[(size_t)row * N + i];
    float y = (v - mean) * rstd * g[i] + b[i];
    if (post == POST_LEAKY)     y = (y >= 0.0f) ? y : 0.01f * y;
    else if (post == POST_GELU) y = 0.5f * y * (1.0f + erff(y * 0.70710678118654752f));
    Y[(size_t)row * N + i] = y;
  }
}

// ---------------------------------------------------------------------------
// MoE gating: softmax over 4 experts, top-2, renormalized combine weights.
// One 64-thread block per row.
// ---------------------------------------------------------------------------
__global__ void moe_gate_kernel(const float* __restrict__ X,
                                const float* __restrict__ G,
                                float* __restrict__ comb, int d) {
  const int r = blockIdx.x;
  const int t = threadIdx.x;
  __shared__ float part[64][4];
  float acc[4] = {0.f, 0.f, 0.f, 0.f};
  for (int i = t; i < d; i += 64) {
    float x = X[(size_t)r * d + i];
    acc[0] += x * G[i * 4 + 0];
    acc[1] += x * G[i * 4 + 1];
    acc[2] += x * G[i * 4 + 2];
    acc[3] += x * G[i * 4 + 3];
  }
  part[t][0] = acc[0]; part[t][1] = acc[1]; part[t][2] = acc[2]; part[t][3] = acc[3];
  __syncthreads();
  if (t == 0) {
    float gsum[4] = {0.f, 0.f, 0.f, 0.f};
    for (int i = 0; i < 64; ++i)
      for (int e = 0; e < 4; ++e) gsum[e] += part[i][e];
    float mx = gsum[0];
    for (int e = 1; e < 4; ++e) mx = fmaxf(mx, gsum[e]);
    float den = 0.0f, p[4];
    for (int e = 0; e < 4; ++e) { p[e] = expf(gsum[e] - mx); den += p[e]; }
    for (int e = 0; e < 4; ++e) p[e] /= den;
    int i1 = 0;
    for (int e = 1; e < 4; ++e) if (p[e] > p[i1]) i1 = e;
    int i2 = -1;
    for (int e = 0; e < 4; ++e) if (e != i1 && (i2 < 0 || p[e] > p[i2])) i2 = e;
    float s = p[i1] + p[i2];
    float w[4] = {0.f, 0.f, 0.f, 0.f};
    w[i1] = p[i1] / s; w[i2] = p[i2] / s;
    for (int e = 0; e < 4; ++e) comb[(size_t)r * 4 + e] = w[e];
  }
}

// ---------------------------------------------------------------------------
// pmask: mask==0 -> masked; if a whole row masked, unmask position 0.
// ---------------------------------------------------------------------------
__global__ void pmask_kernel(const float* __restrict__ mask, int* __restrict__ pm) {
  const int b = blockIdx.x;
  const int t = threadIdx.x;
  __shared__ int f[64];
  int m = (mask[(size_t)b * 64 + t] == 0.0f) ? 1 : 0;
  f[t] = m;
  __syncthreads();
  __shared__ int all;
  if (t == 0) {
    int a = 1;
    for (int i = 0; i < 64; ++i) a &= f[i];
    all = a;
  }
  __syncthreads();
  if (t == 0 && all) m = 0;
  pm[(size_t)b * 64 + t] = m;
}

// ---------------------------------------------------------------------------
// Build context-feature input columns 128..447 (level + cards embeddings).
// Grid 512 blocks x 320 threads.
// ---------------------------------------------------------------------------
__global__ void gather_cin_kernel(const float* __restrict__ context,
                                  const float* __restrict__ level_emb,
                                  const float* __restrict__ cards_emb,
                                  float* __restrict__ cin) {
  const int b = blockIdx.x;
  const int t = threadIdx.x;
  if (t < 128) {
    int lvl = (int)context[(size_t)b * 112 + 108];
    cin[(size_t)b * 448 + 128 + t] = level_emb[(size_t)lvl * 128 + t];
  } else {
    int j = t - 128;              // 0..191
    int slot = j >> 6;            // 0..2
    int i = j & 63;
    int card = (int)context[(size_t)b * 112 + 109 + slot];
    cin[(size_t)b * 448 + 256 + j] = cards_emb[(size_t)card * 64 + i];
  }
}

// ---------------------------------------------------------------------------
// In-place RoPE on q and k halves of the qkv buffer (B,L,1536).
// Thread handles one rotation pair (d, d+32). part: 0=q, 1=k.
// ---------------------------------------------------------------------------
__global__ void rope_kernel(float* __restrict__ qkv) {
  const unsigned idx = blockIdx.x * blockDim.x + threadIdx.x; // B*L*2*8*32
  const int j    = idx & 31;
  const int h    = (idx >> 5) & 7;
  const int part = (idx >> 8) & 1;
  const int l    = (idx >> 9) & 63;
  const int b    = idx >> 15;
  const float ang = (float)l * expf(-(float)j * (9.210340371976184f / 32.0f));
  const float c = cosf(ang), s = sinf(ang);
  size_t base = ((size_t)b * 64 + l) * 1536 + part * 512 + h * 64;
  float x1 = qkv[base + j];
  float x2 = qkv[base + j + 32];
  qkv[base + j]      = x1 * c - x2 * s;
  qkv[base + j + 32] = x2 * c + x1 * s;
}

// ---------------------------------------------------------------------------
// Self-attention: one 64-thread block per (b,h,l).
// Reads q/k/v straight from the (rope'd) qkv buffer.
// ---------------------------------------------------------------------------
__global__ void attn_kernel(const float* __restrict__ qkv,
                            const int* __restrict__ pm,
                            float* __restrict__ o) {
  const int l = blockIdx.x, h = blockIdx.y, b = blockIdx.z;
  const int t = threadIdx.x;
  __shared__ float sq[64], sp[64];
  __shared__ float rinv;
  const size_t rowbase = (size_t)b * 64 * 1536;
  sq[t] = qkv[rowbase + (size_t)l * 1536 + h * 64 + t];
  __syncthreads();
  // score for key m = t
  const float* kb = qkv + rowbase + 512 + h * 64;
  float dot = 0.0f;
  for (int d = 0; d < 64; ++d) dot += sq[d] * kb[(size_t)t * 1536 + d];
  sp[t] = pm[(size_t)b * 64 + t] ? -1e30f : dot * 0.125f;
  __syncthreads();
  if (t == 0) {
    float mx = sp[0];
    for (int m = 1; m < 64; ++m) mx = fmaxf(mx, sp[m]);
    float sm = 0.0f;
    for (int m = 0; m < 64; ++m) { float e = expf(sp[m] - mx); sp[m] = e; sm += e; }
    rinv = 1.0f / sm;
  }
  __syncthreads();
  const float* vb = qkv + rowbase + 1024 + h * 64 + t;
  float accv = 0.0f;
  for (int m = 0; m < 64; ++m) accv += sp[m] * vb[(size_t)m * 1536];
  o[((size_t)b * 64 + l) * 512 + h * 64 + t] = accv * rinv;
}

// ---------------------------------------------------------------------------
// Cross-attention: one 64-thread block per (b,h).
// ---------------------------------------------------------------------------
__global__ void xattn_kernel(const float* __restrict__ qx,
                             const float* __restrict__ kx,
                             const float* __restrict__ vx,
                             const int* __restrict__ pm,
                             float* __restrict__ o2) {
  const int h = blockIdx.x, b = blockIdx.y;
  const int t = threadIdx.x;
  __shared__ float sq[64], sp[64];
  __shared__ float rinv;
  sq[t] = qx[(size_t)b * 512 + h * 64 + t];
  __syncthreads();
  const float* kb = kx + ((size_t)b * 64 + t) * 512 + h * 64;
  float dot = 0.0f;
  for (int d = 0; d < 64; ++d) dot += sq[d] * kb[d];
  sp[t] = pm[(size_t)b * 64 + t] ? -1e30f : dot * 0.125f;
  __syncthreads();
  if (t == 0) {
    float mx = sp[0];
    for (int m = 1; m < 64; ++m) mx = fmaxf(mx, sp[m]);
    float sm = 0.0f;
    for (int m = 0; m < 64; ++m) { float e = expf(sp[m] - mx); sp[m] = e; sm += e; }
    rinv = 1.0f / sm;
  }
  __syncthreads();
  float accv = 0.0f;
  for (int m = 0; m < 64; ++m)
    accv += sp[m] * vx[((size_t)b * 64 + m) * 512 + h * 64 + t];
  o2[(size_t)b * 512 + h * 64 + t] = accv * rinv;
}

// ---------------------------------------------------------------------------
// Small utility kernels.
// ---------------------------------------------------------------------------
__global__ void fill_kernel(float* __restrict__ p, size_t n, float v) {
  size_t i = (size_t)blockIdx.x * blockDim.x + threadIdx.x;
  size_t stride = (size_t)gridDim.x * blockDim.x;
  for (; i < n; i += stride) p[i] = v;
}

__global__ void add_kernel(const float* __restrict__ a, const float* __restrict__ b,
                           float* __restrict__ c, size_t n) {
  size_t i = (size_t)blockIdx.x * blockDim.x + threadIdx.x;
  size_t stride = (size_t)gridDim.x * blockDim.x;
  for (; i < n; i += stride) c[i] = a[i] + b[i];
}

// out[r] = dot(f3[r], W3) + b3
__global__ void final_kernel(const float* __restrict__ f3,
                             const float* __restrict__ W3,
                             const float* __restrict__ b3,
                             float* __restrict__ out) {
  const int r = blockIdx.x;
  const int t = threadIdx.x;
  __shared__ float red[64];
  float acc = 0.0f;
  for (int d = t; d < 512; d += 64) acc += f3[(size_t)r * 512 + d] * W3[d];
  red[t] = acc;
  __syncthreads();
  for (int s = 32; s > 0; s >>= 1) {
    if (t < s) red[t] += red[t + s];
    __syncthreads();
  }
  if (t == 0) out[r] = red[0] + b3[0];
}

// ---------------------------------------------------------------------------
// Host-side helpers
// ---------------------------------------------------------------------------
static void gemm(hipStream_t s, const float* A, int lda, const float* W, int ldb,
                 const float* bias, const float* resid, int ldr,
                 const float* rs, int rss, float* C, int ldc,
                 int M, int N, int K, int mode) {
  dim3 g(N / 64, M / 16);   // all N in this model are multiples of 64
  wmma_gemm_kernel<<<g, 128, 0, s>>>(A, lda, W, ldb, bias, resid, ldr, rs, rss,
                                     C, ldc, K, mode);
}

static void ln(hipStream_t s, const float* X, const float* R_, const float* g,
               const float* b, float* Y, int rows, int N, int post) {
  ln_kernel<<<rows, 256, 0, s>>>(X, R_, g, b, Y, N, post);
}

// Dense MoE block (matches reference: compute all experts, weighted combine,
// LN(out + x)).  d = model dim, hdn = 2d.
static void moe_block(hipStream_t s, const float* x, float* out,
                      float* Hbuf, float* Ybuf, float* comb, int R, int d,
                      const float* gate, const float* W1, const float* b1,
                      const float* g1, const float* bb1,
                      const float* W2, const float* b2,
                      const float* g, const float* b) {
  const int hdn = 2 * d;
  moe_gate_kernel<<<R, 64, 0, s>>>(x, gate, comb, d);
  fill_kernel<<<1024, 256, 0, s>>>(Ybuf, (size_t)R * d, 0.0f);
  for (int e = 0; e < 4; ++e) {
    const float* W1e = W1 + (size_t)e * d * hdn;
    const float* W2e = W2 + (size_t)e * hdn * d;
    gemm(s, x, d, W1e, hdn, b1 + (size_t)e * hdn, nullptr, 0, nullptr, 0,
         Hbuf, hdn, R, hdn, d, GM_BIAS);
    ln(s, Hbuf, nullptr, g1 + (size_t)e * hdn, bb1 + (size_t)e * hdn,
       Hbuf, R, hdn, POST_GELU);
    gemm(s, Hbuf, hdn, W2e, d, b2 + (size_t)e * d, nullptr, 0,
         comb + e, 4, Ybuf, d, R, d, hdn, GM_SCALEACC);
  }
  ln(s, Ybuf, x, g, b, out, R, d, POST_NONE);
}

// ---------------------------------------------------------------------------
// kernel_launch
// ---------------------------------------------------------------------------
extern "C" void kernel_launch(void* const* d_in, const int* in_sizes, int n_in,
                              void* d_out, int out_size, void* d_ws, size_t ws_size,
                              hipStream_t stream) {
  (void)in_sizes; (void)n_in; (void)out_size; (void)ws_size;
  auto P = [&](int i) -> const float* { return (const float*)d_in[i]; };

  // ---- input leaves (setup_inputs insertion order) ----
  const float* query    = P(0);   // (512,216)
  const float* context  = P(1);   // (512,112)
  const float* history  = P(2);   // (512,64,112)
  const float* hmask    = P(3);   // (512,64)
  const float* qe_W = P(4);  const float* qe_b = P(5);
  const float* qe_g = P(6);  const float* qe_beta = P(7);
  // qe_moe: 8..16
  const float* unseen_W = P(17); const float* unseen_b = P(18);
  const float* level_emb = P(19); const float* cards_emb = P(20);
  const float* cf_W = P(21); const float* cf_b = P(22);
  const float* cf_g = P(23); const float* cf_beta = P(24);
  // cf_moe: 25..33
  const float* hp_W = P(34); const float* hp_b = P(35);
  const float* ha_W = P(36); const float* ha_b = P(37);
  const float* hproj_W = P(38); const float* hproj_b = P(39);
  const float* hproj_g = P(40); const float* hproj_beta = P(41);
  const float* qkv_W = P(42); const float* qkv_b = P(43);
  const float* proj_W = P(44); const float* proj_b = P(45);
  const float* attn_g = P(46); const float* attn_b = P(47);
  // hist_moe: 48..56
  const float* xin_W = P(57); const float* xin_b = P(58);
  const float* xout_W = P(59); const float* xout_b = P(60);
  const float* aln_g = P(61); const float* aln_b = P(62);
  const float* fn_W1 = P(63); const float* fn_b1 = P(64);
  const float* fn_g1 = P(65); const float* fn_beta1 = P(66);
  // fn.moe: 67..75
  const float* fn_W2 = P(76); const float* fn_b2 = P(77);
  const float* fn_W3 = P(78); const float* fn_b3 = P(79);

  float* out = (float*)d_out;
  float* ws = (float*)d_ws;

  // ---- workspace arena (floats) ----
  const size_t BL = 512 * 64;  // 32768
  size_t off = 0;
  auto A = [&](size_t n) { size_t r = off; off += n; return r; };
  const size_t o_q     = A(512 * 512);
  const size_t o_qfeat = A(512 * 512);   // qfeat|cfeat|attnout contiguous = f
  const size_t o_cfeat = A(512 * 512);
  const size_t o_aout  = A(512 * 512);
  const size_t o_cin   = A(512 * 448);
  const size_t o_c     = A(512 * 512);
  const size_t o_state = A(512 * 512);
  const size_t o_qx    = A(512 * 512);
  const size_t o_o2    = A(512 * 512);
  const size_t o_mH    = A(512 * 2048);
  const size_t o_mY    = A(512 * 1024);
  const size_t o_combS = A(512 * 4);
  const size_t o_f1    = A(512 * 1024);
  const size_t o_f2    = A(512 * 1024);
  const size_t o_f3    = A(512 * 512);
  const size_t o_pm    = A(512 * 64);    // ints
  const size_t o_hin   = A(BL * 256);
  const size_t o_h     = A(BL * 512);
  const size_t o_qkv   = A(BL * 1536);
  const size_t o_obuf  = A(BL * 512);
  const size_t o_h2    = A(BL * 512);
  const size_t o_combH = A(BL * 4);

  float* q     = ws + o_q;     float* qfeat = ws + o_qfeat;
  float* cfeat = ws + o_cfeat; float* aout  = ws + o_aout;
  float* cin   = ws + o_cin;   float* c     = ws + o_c;
  float* state = ws + o_state; float* qx    = ws + o_qx;
  float* o2    = ws + o_o2;    float* mH    = ws + o_mH;
  float* mY    = ws + o_mY;    float* combS = ws + o_combS;
  float* f1    = ws + o_f1;    float* f2    = ws + o_f2;
  float* f3    = ws + o_f3;    int*   pm    = (int*)(ws + o_pm);
  float* hin   = ws + o_hin;   float* h     = ws + o_h;
  float* qkv   = ws + o_qkv;   float* obuf  = ws + o_obuf;
  float* h2    = ws + o_h2;    float* combH = ws + o_combH;
  // reuse of dead buffers:
  float* hmoeH = qkv;                    // 32768x1024 fits in qkv (dead post-attn)
  float* hmoeY = obuf;                   // dead after proj
  float* kx    = qkv;                    // 32768x512 (dead after hist_moe)
  float* vx    = qkv + BL * 512;         // 32768x512

  // ---- 1. query encoder: q = leaky(LN(query @ qe_W + qe_b)) ----
  gemm(stream, query, 216, qe_W, 512, qe_b, nullptr, 0, nullptr, 0,
       q, 512, 512, 512, 216, GM_BIAS);
  ln(stream, q, nullptr, qe_g, qe_beta, q, 512, 512, POST_LEAKY);

  // ---- 2. q_feat = moe(q, qe_moe) ----
  moe_block(stream, q, qfeat, mH, mY, combS, 512, 512,
            P(8), P(9), P(10), P(11), P(12), P(13), P(14), P(15), P(16));

  // ---- 3. context features ----
  gemm(stream, context, 112, unseen_W, 128, unseen_b, nullptr, 0, nullptr, 0,
       cin, 448, 512, 128, 108, GM_LEAKY);                 // u -> cin[:, :128]
  gather_cin_kernel<<<512, 320, 0, stream>>>(context, level_emb, cards_emb, cin);
  gemm(stream, cin, 448, cf_W, 512, cf_b, nullptr, 0, nullptr, 0,
       c, 512, 512, 512, 448, GM_BIAS);
  ln(stream, c, nullptr, cf_g, cf_beta, c, 512, 512, POST_NONE);
  moe_block(stream, c, cfeat, mH, mY, combS, 512, 512,
            P(25), P(26), P(27), P(28), P(29), P(30), P(31), P(32), P(33));

  // ---- 4. history projection ----
  gemm(stream, history, 112, hp_W, 64, hp_b, nullptr, 0, nullptr, 0,
       hin, 256, 32768, 64, 4, GM_LEAKY);                  // p -> hin[:, :64]
  gemm(stream, history + 4, 112, ha_W, 192, ha_b, nullptr, 0, nullptr, 0,
       hin + 64, 256, 32768, 192, 108, GM_LEAKY);          // a -> hin[:, 64:]
  gemm(stream, hin, 256, hproj_W, 512, hproj_b, nullptr, 0, nullptr, 0,
       h, 512, 32768, 512, 256, GM_BIAS);
  ln(stream, h, nullptr, hproj_g, hproj_beta, h, 32768, 512, POST_NONE);

  // ---- 5. pmask ----
  pmask_kernel<<<512, 64, 0, stream>>>(hmask, pm);

  // ---- 6. self-attention ----
  gemm(stream, h, 512, qkv_W, 1536, qkv_b, nullptr, 0, nullptr, 0,
       qkv, 1536, 32768, 1536, 512, GM_BIAS);
  rope_kernel<<<(512u * 64 * 2 * 8 * 32) / 256, 256, 0, stream>>>(qkv);
  attn_kernel<<<dim3(64, 8, 512), 64, 0, stream>>>(qkv, pm, obuf);
  gemm(stream, obuf, 512, proj_W, 512, proj_b, h, 512, nullptr, 0,
       h2, 512, 32768, 512, 512, GM_RESID);
  ln(stream, h2, nullptr, attn_g, attn_b, h2, 32768, 512, POST_NONE);

  // ---- 7. hist_moe over all 32768 rows (writes into h; qkv/obuf reused) ----
  moe_block(stream, h2, h, hmoeH, hmoeY, combH, 32768, 512,
            P(48), P(49), P(50), P(51), P(52), P(53), P(54), P(55), P(56));

  // ---- 8/9. cross attention ----
  add_kernel<<<512, 256, 0, stream>>>(qfeat, cfeat, state, (size_t)512 * 512);
  gemm(stream, state, 512, xin_W, 1536, xin_b, nullptr, 0, nullptr, 0,
       qx, 512, 512, 512, 512, GM_BIAS);                          // Wq slice
  gemm(stream, h, 512, xin_W + 512, 1536, xin_b + 512, nullptr, 0, nullptr, 0,
       kx, 512, 32768, 512, 512, GM_BIAS);                        // Wk slice
  gemm(stream, h, 512, xin_W + 1024, 1536, xin_b + 1024, nullptr, 0, nullptr, 0,
       vx, 512, 32768, 512, 512, GM_BIAS);                        // Wv slice
  xattn_kernel<<<dim3(8, 512), 64, 0, stream>>>(qx, kx, vx, pm, o2);
  gemm(stream, o2, 512, xout_W, 512, xout_b, state, 512, nullptr, 0,
       aout, 512, 512, 512, 512, GM_RESID);
  ln(stream, aout, nullptr, aln_g, aln_b, aout, 512, 512, POST_NONE);

  // ---- 10. final head: f = [qfeat|cfeat|aout] is contiguous at qfeat ----
  gemm(stream, qfeat, 1536, fn_W1, 1024, fn_b1, nullptr, 0, nullptr, 0,
       f1, 1024, 512, 1024, 1536, GM_BIAS);
  ln(stream, f1, nullptr, fn_g1, fn_beta1, f1, 512, 1024, POST_NONE);
  moe_block(stream, f1, f2, mH, mY, combS, 512, 1024,
            P(67), P(68), P(69), P(70), P(71), P(72), P(73), P(74), P(75));
  gemm(stream, f2, 1024, fn_W2, 512, fn_b2, nullptr, 0, nullptr, 0,
       f3, 512, 512, 512, 1024, GM_LEAKY);
  final_kernel<<<512, 64, 0, stream>>>(f3, fn_W3, fn_b3, out);
}